// TracePooling_77171972375022
// MI455X (gfx1250) — compile-verified
//
#include <hip/hip_runtime.h>
#include <hip/hip_bf16.h>

typedef __attribute__((ext_vector_type(2))) float v2f;
typedef __attribute__((ext_vector_type(8))) float v8f;

__device__ __forceinline__ float rdlane_f(float v, int l) {
  return __uint_as_float(__builtin_amdgcn_readlane(__float_as_uint(v), l));
}

template <int OFF>
__device__ __forceinline__ float swz_f(float v) {
  return __uint_as_float(
      (unsigned)__builtin_amdgcn_ds_swizzle((int)__float_as_uint(v), OFF));
}

// One elimination step of unpivoted LU on a 16x16 matrix held in WMMA C-layout.
//   c : A in C-layout        (c[r] @ lane l = A[r + 8*(l>=16), l%16])
//   t : A^T in C-layout      (t[r] @ lane l = A[l%16, r + 8*(l>=16)])
// Trailing update done as D = C + (-col_k/akk) x row_k via V_WMMA_F32_16X16X4_F32
// (only the K=0 slice of A/B operands is populated; rows <= k of the A operand
// are zero so already-factored rows are preserved). A second, transposed WMMA
// keeps the mirror `t` consistent so column extraction stays a single swizzle.
template <int K>
__device__ __forceinline__ void lu_step(v8f& c, v8f& t, float& det, int lane) {
  constexpr int half   = (K >= 8) ? 1 : 0;
  constexpr int pl     = K + 16 * half;                 // lane holding A[K][K]
  constexpr int swzoff = ((half * 16) << 5) | 0x0F;     // src = (l & 0xF) | 16*half

  float akk = rdlane_f(c[K & 7], pl);
  det *= akk;                                           // det = prod of pivots

  if constexpr (K < 15) {
    const int  n     = lane & 15;
    const bool lo    = lane < 16;
    const bool below = lo && (n > K);

    float negr = -__builtin_amdgcn_rcpf(akk);
    float rowk = swz_f<swzoff>(c[K & 7]);               // A[K][n%16] at every lane
    float colk = swz_f<swzoff>(t[K & 7]);               // A[n%16][K] at every lane
    float ml   = colk * negr;                           // -multiplier per row

    // c += (-L_col masked rows>K) x (U_row)     : 16x16x4, K=0 slice only
    v2f a1; a1.x = below ? ml   : 0.0f; a1.y = 0.0f;
    v2f b1; b1.x = lo    ? rowk : 0.0f; b1.y = 0.0f;
    c = __builtin_amdgcn_wmma_f32_16x16x4_f32(false, a1, false, b1,
                                              (short)0, c, false, false);
    // t += (U_row as M-vector) x (-L_col masked cols>K)
    v2f a2; a2.x = lo    ? rowk : 0.0f; a2.y = 0.0f;
    v2f b2; b2.x = below ? ml   : 0.0f; b2.y = 0.0f;
    t = __builtin_amdgcn_wmma_f32_16x16x4_f32(false, a2, false, b2,
                                              (short)0, t, false, false);
  }
}

template <int K>
__device__ __forceinline__ void lu_all(v8f& c, v8f& t, float& det, int lane) {
  lu_step<K>(c, t, det, lane);
  if constexpr (K < 15) lu_all<K + 1>(c, t, det, lane);
}

__device__ __forceinline__ void load16(const float* __restrict__ m, int lane,
                                       v8f& c, v8f& t) {
  const int n  = lane & 15;
  const int hi = (lane >> 4) << 3;  // 0 or 8
#pragma unroll
  for (int r = 0; r < 8; ++r) {
    c[r] = m[(r + hi) * 16 + n];    // coalesced: lanes 0..15 read a full row
    t[r] = m[n * 16 + (r + hi)];    // transposed; served from cache (same 1KB)
  }
}

__global__ __launch_bounds__(256) void slater_det16_kernel(
    const float* __restrict__ up, const float* __restrict__ dn,
    float* __restrict__ out, int npairs) {
  const int lane = (int)(threadIdx.x & 31u);
  const int wid  = (int)((blockIdx.x * blockDim.x + threadIdx.x) >> 5);
  if (wid >= npairs) return;  // wave-uniform; EXEC stays all-ones for WMMA

  const float* mu = up + (size_t)wid * 256;
  const float* md = dn + (size_t)wid * 256;
  __builtin_prefetch(md, 0, 0);   // global_prefetch_b8: pull down-matrix early

  v8f cu, tu, cd, td;
  load16(mu, lane, cu, tu);
  load16(md, lane, cd, td);       // both in flight before the WMMA chain

  float du = 1.0f, dd = 1.0f;
  lu_all<0>(cu, tu, du, lane);
  lu_all<0>(cd, td, dd, lane);

  if (lane == 0) out[wid] = du * dd;
}

extern "C" void kernel_launch(void* const* d_in, const int* in_sizes, int n_in,
                              void* d_out, int out_size, void* d_ws, size_t ws_size,
                              hipStream_t stream) {
  const float* up = (const float*)d_in[0];
  const float* dn = (const float*)d_in[1];
  float* out = (float*)d_out;

  const int npairs = in_sizes[0] / 256;        // 4096*64 = 262144 matrices
  const int threads = 256;                     // 8 wave32 per block
  const int waves_per_block = threads / 32;
  const int blocks = (npairs + waves_per_block - 1) / waves_per_block;

  slater_det16_kernel<<<blocks, threads, 0, stream>>>(up, dn, out, npairs);
}